// OuterProduct_6227702579223
// MI455X (gfx1250) — compile-verified
//
#include <hip/hip_runtime.h>

typedef __attribute__((ext_vector_type(2))) float    v2f;
typedef __attribute__((ext_vector_type(8))) float    v8f;
typedef __attribute__((ext_vector_type(4))) unsigned v4u;
typedef __attribute__((ext_vector_type(8))) unsigned v8u;

#define F_FEATS 32
#define DDIM    64
#define NPAIR   496
#define BATCHSZ 4096
#define WAVES_PER_BLOCK 8
#define BLOCK_THREADS   256
#define M_PER_WAVE      32                      // two 16-row M tiles per wave
#define BATCH_PER_BLOCK (WAVES_PER_BLOCK * M_PER_WAVE)   // 256

// LDS strides chosen for bank-conflict freedom (64 banks x 4B):
//  - xi rows padded to 68 floats (TDM pad: 4 DWORDs every 64 DWORDs): row delta-bank = 4
//    -> the 32 lanes' b64 A-reads tile all 64 banks exactly
//  - W pair-rows padded to 80 float2: row delta-bank = 32 -> lane halves hit disjoint bank halves
#define XI_STRIDE 68
#define WP_STRIDE 80

#define SW_FLOATS  (32 * WP_STRIDE * 2)                        // 5120 floats = 20 KB
#define SXI_FLOATS (BATCH_PER_BLOCK * XI_STRIDE)               // 17408 floats = 68 KB

// out[b, p] = sum_{d,e} x[b,i_p,d] * W[p,d,e] * x[b,j_p,e]
__global__ __launch_bounds__(BLOCK_THREADS)
void bilinear_pairs_wmma_f32(const float* __restrict__ x,   // [B, F, D]
                             const float* __restrict__ W,   // [P, D, D]
                             float* __restrict__ out)       // [B, P]
{
    // Single carve-out: sW at LDS offset 0 (B-loads get 16-bit immediates),
    // sXi at byte offset SW_FLOATS*4 = 20480 (TDM destination).
    __shared__ __align__(16) float smem[SW_FLOATS + SXI_FLOATS];   // 88 KB
    float* const sW  = smem;                 // paired W: sW[(q*WP_STRIDE+e)*2 + par] = W[2q+par][e]
    float* const sXi = smem + SW_FLOATS;     // 256 x 64 xi tile, rows padded to XI_STRIDE by TDM

    const int p      = blockIdx.x;
    const int bblock = blockIdx.y;
    const int tid    = threadIdx.x;
    const int wave   = tid >> 5;
    const int lane   = tid & 31;
    const int half   = lane >> 4;   // which 16-lane half
    const int lhalf  = lane & 15;

    // pair index -> (i, j) upper-triangular (k=1); uniform scalar loop
    int i = 0, rem = p;
    while (rem >= F_FEATS - 1 - i) { rem -= F_FEATS - 1 - i; ++i; }
    const int j = i + 1 + rem;

    const int b0 = bblock * BATCH_PER_BLOCK + wave * M_PER_WAVE;

    // ---- TDM: async-load the whole block's Xi tile (256 rows x 64 f32) into LDS ----
    // 2D descriptor: tile 64 x 256, global row stride = F*D floats, data_size = 4B,
    // LDS pad 4 DWORDs every 64 DWORDs -> effective LDS row stride = XI_STRIDE floats.
    if (wave == 0) {
        const unsigned long long ga =
            (unsigned long long)(uintptr_t)(x + ((size_t)bblock * BATCH_PER_BLOCK * F_FEATS + i) * DDIM);

        v4u g0 = {};
        g0[0] = 1u;                                            // count=1 (valid user D#)
        g0[1] = (unsigned)(SW_FLOATS * 4);                     // lds_addr (bytes): sXi offset
        g0[2] = (unsigned)(ga & 0xFFFFFFFFu);                  // global_addr[31:0]
        g0[3] = (unsigned)((ga >> 32) & 0x01FFFFFFu)           // global_addr[56:32]
              | (2u << 30);                                    // type=2 ("image")

        v8u g1 = {};
        g1[0] = (2u << 16)                                     // data_size: 4 bytes
              | (1u << 20)                                     // pad_enable
              | (5u << 22)                                     // pad_interval: 64 DWORDs
              | (3u << 25);                                    // pad_amount: 4 DWORDs
        g1[1] = (unsigned)(DDIM & 0xFFFF) << 16;               // tensor_dim0[15:0]=64
        g1[2] = (unsigned)(BATCH_PER_BLOCK & 0xFFFF) << 16;    // tensor_dim0[31:16]=0; tensor_dim1[15:0]=256
        g1[3] = (unsigned)(DDIM & 0xFFFF) << 16;               // tensor_dim1[31:16]=0; tile_dim0=64
        g1[4] = (unsigned)BATCH_PER_BLOCK;                     // tile_dim1=256; tile_dim2=0 (2D)
        g1[5] = (unsigned)(F_FEATS * DDIM);                    // tensor_dim0_stride=2048 (elements)
        g1[6] = 0u;                                            // stride[47:32]=0; tensor_dim1_stride lo=0
        g1[7] = 0u;

        // 2-group form (VADDR2/VADDR3 = NULL); wait in the same asm to pin ordering.
        asm volatile("tensor_load_to_lds %0, %1\n\t"
                     "s_wait_tensorcnt 0x0"
                     :: "s"(g0), "s"(g1) : "memory");
    }

    // ---- Stage W_p into LDS in paired-row layout (4x float4 per thread) ----
    {
        const float4* Wp = (const float4*)(W + (size_t)p * (DDIM * DDIM));
        #pragma unroll
        for (int k = 0; k < 4; ++k) {
            const int idx = tid + k * BLOCK_THREADS;  // float4 index, 1024 total
            const float4 w = Wp[idx];
            const int d  = idx >> 4;          // row of W_p
            const int e0 = (idx & 15) * 4;    // first of 4 consecutive e
            float* base = sW + ((d >> 1) * WP_STRIDE + e0) * 2 + (d & 1);
            base[0] = w.x; base[2] = w.y; base[4] = w.z; base[6] = w.w;
        }
    }
    __syncthreads();

    // ---- L = Xi (32x64) @ W_p (64x64) via v_wmma_f32_16x16x4_f32 ----
    // Per-lane, loop-invariant bases; all per-(k,nt,mt) deltas are compile-time immediates.
    const float* const xiw   = sXi + wave * (M_PER_WAVE * XI_STRIDE);
    const float* const wbase = sW + (half * WP_STRIDE + lhalf) * 2;          // B base
    const float* const abase = xiw + lhalf * XI_STRIDE + 2 * half;           // A base

    v8f acc[2][4] = {};

    #pragma unroll
    for (int k = 0; k < 16; ++k) {
        // A fragments: {xi[M, d0], xi[M, d0+1]}, M = lhalf (+16)
        const v2f a0 = *(const v2f*)(abase + 4 * k);
        const v2f a1 = *(const v2f*)(abase + 16 * XI_STRIDE + 4 * k);

        // B fragments: paired layout -> single aligned ds_load_b64 each
        const v2f b0f = *(const v2f*)(wbase + (k * 2 * WP_STRIDE +  0) * 2);
        const v2f b1f = *(const v2f*)(wbase + (k * 2 * WP_STRIDE + 16) * 2);
        const v2f b2f = *(const v2f*)(wbase + (k * 2 * WP_STRIDE + 32) * 2);
        const v2f b3f = *(const v2f*)(wbase + (k * 2 * WP_STRIDE + 48) * 2);

        acc[0][0] = __builtin_amdgcn_wmma_f32_16x16x4_f32(false, a0, false, b0f, (short)0, acc[0][0], false, false);
        acc[0][1] = __builtin_amdgcn_wmma_f32_16x16x4_f32(false, a0, false, b1f, (short)0, acc[0][1], false, false);
        acc[0][2] = __builtin_amdgcn_wmma_f32_16x16x4_f32(false, a0, false, b2f, (short)0, acc[0][2], false, false);
        acc[0][3] = __builtin_amdgcn_wmma_f32_16x16x4_f32(false, a0, false, b3f, (short)0, acc[0][3], false, false);
        acc[1][0] = __builtin_amdgcn_wmma_f32_16x16x4_f32(false, a1, false, b0f, (short)0, acc[1][0], false, false);
        acc[1][1] = __builtin_amdgcn_wmma_f32_16x16x4_f32(false, a1, false, b1f, (short)0, acc[1][1], false, false);
        acc[1][2] = __builtin_amdgcn_wmma_f32_16x16x4_f32(false, a1, false, b2f, (short)0, acc[1][2], false, false);
        acc[1][3] = __builtin_amdgcn_wmma_f32_16x16x4_f32(false, a1, false, b3f, (short)0, acc[1][3], false, false);
    }

    // ---- Epilogue: out[b, p] = sum_e L[b, e] * xj[b, e] ----
    // D layout: VGPR r, lane-half h, lane n=lhalf -> L[tile_base + r + 8h, 16*nt + n]
    const float* const xjbase  = x + ((size_t)(b0 + 8 * half) * F_FEATS + j) * DDIM + lhalf;
    float* const       outbase = out + (size_t)(b0 + 8 * half) * NPAIR + p;

    #pragma unroll
    for (int mt = 0; mt < 2; ++mt) {
        #pragma unroll
        for (int r = 0; r < 8; ++r) {
            const int brow = mt * 16 + r;                          // offset from (b0 + 8*half)
            const float* xjrow = xjbase + (size_t)brow * (F_FEATS * DDIM);

            float sum = acc[mt][0][r] * xjrow[ 0]
                      + acc[mt][1][r] * xjrow[16]
                      + acc[mt][2][r] * xjrow[32]
                      + acc[mt][3][r] * xjrow[48];

            // reduce across the 16 lanes of this half (wave32; masks stay in-half)
            sum += __shfl_xor(sum, 1);
            sum += __shfl_xor(sum, 2);
            sum += __shfl_xor(sum, 4);
            sum += __shfl_xor(sum, 8);

            if (lhalf == 0)
                outbase[(size_t)brow * NPAIR] = sum;
        }
    }
}

extern "C" void kernel_launch(void* const* d_in, const int* in_sizes, int n_in,
                              void* d_out, int out_size, void* d_ws, size_t ws_size,
                              hipStream_t stream) {
    const float* x = (const float*)d_in[0];   // [4096, 32, 64] f32
    const float* W = (const float*)d_in[1];   // [496, 64, 64] f32
    float* out = (float*)d_out;               // [4096, 496] f32

    dim3 grid(NPAIR, BATCHSZ / BATCH_PER_BLOCK);   // 496 x 16
    bilinear_pairs_wmma_f32<<<grid, dim3(BLOCK_THREADS), 0, stream>>>(x, W, out);
}